// MassConservingLSTM_MR_22591527977660
// MI455X (gfx1250) — compile-verified
//
#include <hip/hip_runtime.h>
#include <math.h>

// MassConservingLSTM-MR persistent-kernel port for MI455X (gfx1250).
// - 32 workgroups x 256 threads (8 wave32 waves), each owns 16 batch rows.
// - All matmuls via V_WMMA_F32_16X16X4_F32 (fp32 precision preserved).
// - c-state + all per-step activations live in LDS (~114 KB dynamic).
// - Cross-WG scalar L1(c) per step via deterministic slot array + rel/acq counter.

#define TT 512
#define BB 512
#define NWG 32
#define ROWS 16
#define GDIM 64
#define ODIM 32

typedef float v2f __attribute__((ext_vector_type(2)));
typedef float v8f __attribute__((ext_vector_type(8)));

// LDS row strides (floats), padded to dodge bank conflicts
#define LD_FEAT 66
#define LD_CNB  34
#define LD_R    1026
#define LD_I    130
#define LD_O    34
#define LD_C    34
#define LD_OV0  66
#define LD_OV1  34
#define LD_M    34
#define LD_EWS  66
#define LD_SWR  34

enum {
  OFF_FEAT = 0,
  OFF_CNB  = OFF_FEAT + ROWS*LD_FEAT,
  OFF_R    = OFF_CNB  + ROWS*LD_CNB,
  OFF_I    = OFF_R    + ROWS*LD_R,
  OFF_O    = OFF_I    + ROWS*LD_I,
  OFF_C    = OFF_O    + ROWS*LD_O,
  OFF_OV0  = OFF_C    + ROWS*LD_C,
  OFF_OV1  = OFF_OV0  + ROWS*LD_OV0,
  OFF_M    = OFF_OV1  + ROWS*LD_OV1,
  OFF_DENR = OFF_M    + ROWS*LD_M,     // reciprocal L1 denoms for r (16x32)
  OFF_DENI = OFF_DENR + ROWS*ODIM,     // reciprocal L1 denoms for i (16x4)
  OFF_EWS  = OFF_DENI + ROWS*4,        // exp(W_s)      [32][66]
  OFF_SWR  = OFF_EWS  + ODIM*LD_EWS,   // sigmoid(W_rm^T)[64][34]
  OFF_MU   = OFF_SWR  + GDIM*LD_SWR,
  OFF_RS   = OFF_MU   + ROWS,
  OFF_RED  = OFF_RS   + ROWS,          // 256-wide deterministic reduction
  OFF_INV  = OFF_RED  + 256,
  SMEM_FLOATS = OFF_INV + 8
};

__device__ __forceinline__ float sigf(float x) { return 1.0f / (1.0f + expf(-x)); }

// A from LDS (feat tile), B from global weight W (row-major [nOut][64], used as W^T)
__device__ __forceinline__ v8f wmma_acc_gB(const float* sA, int lda,
                                           const float* __restrict__ W,
                                           int col0, int nk, int lane) {
  v8f acc = {0.f,0.f,0.f,0.f,0.f,0.f,0.f,0.f};
  const int row  = lane & 15;
  const int half = (lane >> 4) << 1;      // 0 for lanes 0-15, 2 for 16-31
  const int col  = col0 + row;
  for (int ks = 0; ks < nk; ++ks) {
    const int k = ks*4 + half;
    v2f a; a.x = sA[row*lda + k]; a.y = sA[row*lda + k + 1];
    v2f b; b.x = W[col*GDIM + k]; b.y = W[col*GDIM + k + 1];
    acc = __builtin_amdgcn_wmma_f32_16x16x4_f32(false, a, false, b, (short)0, acc,
                                                false, false);
  }
  return acc;
}

// A from LDS, B from LDS (row-major [K][N] with stride ldb)
__device__ __forceinline__ v8f wmma_acc_lB(const float* sA, int lda,
                                           const float* sB, int ldb,
                                           int col0, int nk, int lane) {
  v8f acc = {0.f,0.f,0.f,0.f,0.f,0.f,0.f,0.f};
  const int row  = lane & 15;
  const int half = (lane >> 4) << 1;
  const int col  = col0 + row;
  for (int ks = 0; ks < nk; ++ks) {
    const int k = ks*4 + half;
    v2f a; a.x = sA[row*lda + k];    a.y = sA[row*lda + k + 1];
    v2f b; b.x = sB[k*ldb + col];    b.y = sB[(k+1)*ldb + col];
    acc = __builtin_amdgcn_wmma_f32_16x16x4_f32(false, a, false, b, (short)0, acc,
                                                false, false);
  }
  return acc;
}

// D tile (16x16 f32): VGPR v -> row v (lanes 0-15) / row v+8 (lanes 16-31), col = lane&15
__device__ __forceinline__ void store_tile(v8f acc, float* sOut, int ldo, int col0,
                                           const float* __restrict__ bias, int act,
                                           int lane) {
  const int col   = col0 + (lane & 15);
  const int rbase = (lane >> 4) * 8;
  const float bv = bias ? bias[col] : 0.0f;
#pragma unroll
  for (int v = 0; v < 8; ++v) {
    float x = acc[v] + bv;
    if (act == 1)      x = fmaxf(x, 0.0f);
    else if (act == 2) x = sigf(x);
    sOut[(rbase + v)*ldo + col] = x;
  }
}

__global__ void mclstm_init_sync(int* __restrict__ gcount, float* __restrict__ gpart) {
  int idx = blockIdx.x * blockDim.x + threadIdx.x;
  if (idx < (TT + 1)) gcount[idx] = 0;
  if (idx < (TT + 1) * NWG) gpart[idx] = 0.0f;
}

__global__ __launch_bounds__(256)
void mclstm_persistent(const float* __restrict__ xm, const float* __restrict__ xa,
                       const float* __restrict__ W_i, const float* __restrict__ b_i,
                       const float* __restrict__ W_r, const float* __restrict__ b_r,
                       const float* __restrict__ W_o, const float* __restrict__ b_o,
                       const float* __restrict__ W_s, const float* __restrict__ W_rm,
                       const float* __restrict__ b0, const float* __restrict__ ln_g,
                       const float* __restrict__ ln_b,
                       float* __restrict__ out,
                       int* __restrict__ gcount, float* __restrict__ gpart) {
  extern __shared__ float sm[];
  const int tid  = threadIdx.x;
  const int lane = tid & 31;
  const int w    = tid >> 5;
  const int wg   = blockIdx.x;
  const int rowBase = wg * ROWS;
  const size_t TB32 = (size_t)TT * BB * ODIM;

  // ---- one-time: precompute exp(W_s) [o][g] and sigmoid(W_rm^T) [g][o]; zero c ----
  for (int idx = tid; idx < ODIM * GDIM; idx += 256) {
    const int o = idx >> 6, g = idx & 63;
    sm[OFF_EWS + o*LD_EWS + g] = expf(W_s[idx]);
    sm[OFF_SWR + g*LD_SWR + o] = sigf(W_rm[idx]);
  }
  for (int idx = tid; idx < ROWS * ODIM; idx += 256)
    sm[OFF_C + (idx >> 5)*LD_C + (idx & 31)] = 0.0f;
  __syncthreads();

  for (int t = 0; t < TT; ++t) {
    // ---- phase 1: obtain global inv-norm scalar for this step ----
    if (tid == 0) {
      if (t > 0) {
        while (__hip_atomic_load(&gcount[t], __ATOMIC_ACQUIRE,
                                 __HIP_MEMORY_SCOPE_AGENT) < NWG)
          __builtin_amdgcn_s_sleep(2);
        float s = 0.0f;
        for (int i = 0; i < NWG; ++i)   // fixed order -> deterministic
          s += __hip_atomic_load(&gpart[t*NWG + i], __ATOMIC_RELAXED,
                                 __HIP_MEMORY_SCOPE_AGENT);
        sm[OFF_INV] = 1.0f / (s + 1e-5f);
      } else {
        sm[OFF_INV] = 0.0f;             // c==0 at t==0
      }
    }
    __syncthreads();
    const float inv = sm[OFF_INV];

    // ---- phase 2: build feat [16][64] = [xm | xa | c_norm]; and (c_norm - b0) ----
    for (int idx = tid; idx < ROWS * GDIM; idx += 256) {
      const int r = idx >> 6, col = idx & 63;
      const int b = rowBase + r;
      float v;
      if (col < 4)       v = xm[(size_t)t*BB*4  + b*4  + col];
      else if (col < 32) v = xa[(size_t)t*BB*28 + b*28 + (col - 4)];
      else {
        const int o = col - 32;
        const float cn = sm[OFF_C + r*LD_C + o] * inv;
        sm[OFF_CNB + r*LD_CNB + o] = cn - b0[o];
        v = cn;
      }
      sm[OFF_FEAT + r*LD_FEAT + col] = v;
    }
    __syncthreads();

    // ---- phase 3: all WMMA matmuls (78 16x16 tile jobs over 8 waves) ----
    for (int job = w; job < 78; job += 8) {
      if (job < 64) {               // r = relu(feat @ W_r^T + b_r)  [16][1024]
        v8f acc = wmma_acc_gB(&sm[OFF_FEAT], LD_FEAT, W_r, job*16, 16, lane);
        store_tile(acc, &sm[OFF_R], LD_R, job*16, b_r, 1, lane);
      } else if (job < 72) {        // i = sigmoid(feat @ W_i^T + b_i) [16][128]
        const int n = job - 64;
        v8f acc = wmma_acc_gB(&sm[OFF_FEAT], LD_FEAT, W_i, n*16, 16, lane);
        store_tile(acc, &sm[OFF_I], LD_I, n*16, b_i, 2, lane);
      } else if (job < 74) {        // o = sigmoid(feat @ W_o^T + b_o) [16][32]
        const int n = job - 72;
        v8f acc = wmma_acc_gB(&sm[OFF_FEAT], LD_FEAT, W_o, n*16, 16, lane);
        store_tile(acc, &sm[OFF_O], LD_O, n*16, b_o, 2, lane);
      } else {                      // ov0 = (c_norm - b0) @ exp(W_s)  [16][64]
        const int n = job - 74;
        v8f acc = wmma_acc_lB(&sm[OFF_CNB], LD_CNB, &sm[OFF_EWS], LD_EWS, n*16, 8, lane);
        store_tile(acc, &sm[OFF_OV0], LD_OV0, n*16, nullptr, 0, lane);
      }
    }
    __syncthreads();

    // ---- phase 4: L1 reciprocal denoms for r and i; tanh(ov0) in place ----
    for (int idx = tid; idx < ROWS * ODIM; idx += 256) {
      const int r = idx >> 5, o = idx & 31;
      const float* p = &sm[OFF_R + r*LD_R + o*32];
      float s = 0.0f;
      for (int j = 0; j < 32; ++j) s += p[j];
      sm[OFF_DENR + idx] = 1.0f / fmaxf(s, 1e-12f);
    }
    for (int idx = tid; idx < ROWS * 4; idx += 256) {
      const int r = idx >> 2, ii = idx & 3;
      const float* p = &sm[OFF_I + r*LD_I + ii*32];
      float s = 0.0f;
      for (int j = 0; j < 32; ++j) s += p[j];
      sm[OFF_DENI + idx] = 1.0f / fmaxf(s, 1e-12f);
    }
    for (int idx = tid; idx < ROWS * GDIM; idx += 256) {
      const int r = idx >> 6, g = idx & 63;
      sm[OFF_OV0 + r*LD_OV0 + g] = tanhf(sm[OFF_OV0 + r*LD_OV0 + g]);
    }
    __syncthreads();

    // ---- phase 5: waves 0-1: ov1 = tanh(ov0) @ sig(W_rm^T) (WMMA);
    //               waves 2-7: m_new = xm@i_n + c@r_n ----
    if (w < 2) {
      v8f acc = wmma_acc_lB(&sm[OFF_OV0], LD_OV0, &sm[OFF_SWR], LD_SWR, w*16, 16, lane);
      store_tile(acc, &sm[OFF_OV1], LD_OV1, w*16, nullptr, 0, lane);
    } else {
      for (int idx = tid - 64; idx < ROWS * ODIM; idx += 192) {
        const int r = idx >> 5, p = idx & 31;
        float m = 0.0f;
        for (int ii = 0; ii < 4; ++ii)
          m += sm[OFF_FEAT + r*LD_FEAT + ii] *
               sm[OFF_I + r*LD_I + ii*32 + p] * sm[OFF_DENI + r*4 + ii];
        for (int o = 0; o < ODIM; ++o)
          m += sm[OFF_C + r*LD_C + o] *
               sm[OFF_R + r*LD_R + o*32 + p] * sm[OFF_DENR + r*ODIM + o];
        sm[OFF_M + r*LD_M + p] = m;
      }
    }
    __syncthreads();

    // ---- phase 6a: layer-norm row statistics of ov1 ----
    if (tid < ROWS) {
      const float* p = &sm[OFF_OV1 + tid*LD_OV1];
      float mu = 0.0f;
      for (int j = 0; j < ODIM; ++j) mu += p[j];
      mu *= (1.0f / ODIM);
      float var = 0.0f;
      for (int j = 0; j < ODIM; ++j) { const float d = p[j] - mu; var += d * d; }
      var *= (1.0f / ODIM);
      sm[OFF_MU + tid] = mu;
      sm[OFF_RS + tid] = rsqrtf(var + 1e-5f);
    }
    __syncthreads();

    // ---- phase 6b: MR gate, outputs, c update, local |c_new| partial ----
    float local_abs = 0.0f;
    for (int idx = tid; idx < ROWS * ODIM; idx += 256) {
      const int r = idx >> 5, o = idx & 31;
      const float ov1n = (sm[OFF_OV1 + r*LD_OV1 + o] - sm[OFF_MU + r]) *
                         sm[OFF_RS + r] * ln_g[o] + ln_b[o];
      const float oo = sm[OFF_O + r*LD_O + o];
      const float f  = 1.0f - oo;
      const float ov2 = ov1n - fmaxf(ov1n - f, 0.0f);
      const float MR  = fmaxf(ov2 + 1.0f - f, 0.0f) + f - 1.0f;
      const float m   = sm[OFF_M + r*LD_M + o];
      const float opr  = oo + MR;
      const float h    = oo * m;
      const float cnew = (1.0f - opr) * m;
      const float mrf  = MR * m;
      const size_t off = ((size_t)t * BB + (rowBase + r)) * ODIM + o;
      out[0*TB32 + off] = h;
      out[1*TB32 + off] = cnew;
      out[2*TB32 + off] = oo;
      out[3*TB32 + off] = MR;
      out[4*TB32 + off] = opr;
      out[5*TB32 + off] = mrf;
      out[6*TB32 + off] = h;
      sm[OFF_C + r*LD_C + o] = cnew;
      local_abs += fabsf(cnew);
    }
    // deterministic fixed-order tree reduction of |c_new| partial
    sm[OFF_RED + tid] = local_abs;
    __syncthreads();
    for (int offr = 128; offr > 0; offr >>= 1) {
      if (tid < offr) sm[OFF_RED + tid] += sm[OFF_RED + tid + offr];
      __syncthreads();
    }
    if (tid == 0) {
      __hip_atomic_store(&gpart[(t + 1)*NWG + wg], sm[OFF_RED], __ATOMIC_RELAXED,
                         __HIP_MEMORY_SCOPE_AGENT);
      __hip_atomic_fetch_add(&gcount[t + 1], 1, __ATOMIC_RELEASE,
                             __HIP_MEMORY_SCOPE_AGENT);
    }
    __syncthreads();
  }
}

extern "C" void kernel_launch(void* const* d_in, const int* in_sizes, int n_in,
                              void* d_out, int out_size, void* d_ws, size_t ws_size,
                              hipStream_t stream) {
  const float* xm   = (const float*)d_in[0];
  const float* xa   = (const float*)d_in[1];
  const float* W_i  = (const float*)d_in[2];
  const float* b_i  = (const float*)d_in[3];
  const float* W_r  = (const float*)d_in[4];
  const float* b_r  = (const float*)d_in[5];
  const float* W_o  = (const float*)d_in[6];
  const float* b_o  = (const float*)d_in[7];
  const float* W_s  = (const float*)d_in[8];
  const float* W_rm = (const float*)d_in[9];
  const float* b0   = (const float*)d_in[10];
  const float* ln_g = (const float*)d_in[11];
  const float* ln_b = (const float*)d_in[12];
  float* out = (float*)d_out;

  int*   gcount = (int*)d_ws;
  float* gpart  = (float*)((char*)d_ws + 4096);

  const int initN = (TT + 1) * NWG;
  mclstm_init_sync<<<(initN + 255) / 256, 256, 0, stream>>>(gcount, gpart);
  mclstm_persistent<<<NWG, 256, SMEM_FLOATS * sizeof(float), stream>>>(
      xm, xa, W_i, b_i, W_r, b_r, W_o, b_o, W_s, W_rm, b0, ln_g, ln_b,
      out, gcount, gpart);
}